// CrfDecoder_54829552501304
// MI455X (gfx1250) — compile-verified
//
#include <hip/hip_runtime.h>
#include <hip/hip_bf16.h>
#include <math.h>

typedef __attribute__((ext_vector_type(2))) float v2f;
typedef __attribute__((ext_vector_type(8))) float v8f;

#define TT 1024
#define NN 128
#define STRIDE 132   // 128 + 4 pad: row-to-row bank rotation of 4 banks

// ---------------------------------------------------------------------------
// Forward scan: one workgroup per 16 batch rows (grid = 4). 8 waves per WG,
// each wave owns one 16-column tile of the 16x128 alpha state.
// log-semiring matmul done as exp/GEMM/log with per-row max subtraction.
// B-matrix fragments (exp(transitions)) are loop-invariant and register-resident.
// ---------------------------------------------------------------------------
__global__ __launch_bounds__(256) void crf_scan_kernel(
    const float* __restrict__ emissions,      // [64][1024][1][128]
    const int*   __restrict__ token_sizes,    // [64]
    const float* __restrict__ trans,          // [1][1][128][128]
    const float* __restrict__ head,           // [128]
    const float* __restrict__ last,           // [128]
    float* __restrict__ out)                  // [2][64][1]
{
    __shared__ float alpha_s[16 * STRIDE];
    __shared__ float expA_s [16 * STRIDE];
    __shared__ float red_s  [16 * 16];
    __shared__ float Mrow_s [16];
    __shared__ int   ts_s   [16];

    const int tid  = threadIdx.x;
    const int wave = tid >> 5;        // 0..7 -> column tile
    const int lane = tid & 31;
    const int g    = (lane >> 4) & 1; // lane half
    const int ln   = lane & 15;
    const int b0   = blockIdx.x * 16; // global batch base

    if (tid < 16) ts_s[tid] = token_sizes[b0 + tid];

    // alpha0 = head + emissions[:, 0]
    for (int e = tid; e < 16 * NN; e += 256) {
        const int m = e >> 7, c = e & 127;
        alpha_s[m * STRIDE + c] = head[c] + emissions[(size_t)(b0 + m) * TT * NN + c];
    }
    __syncthreads();

    // Loop-invariant B fragments: B[K][N] = exp(trans[K][N]) for this wave's cols.
    // 16x16x4 f32 B layout: VGPR v, lane-half g holds row K = v + 2g, N = lane&15.
    v2f bfrag[32];
    {
        const int col = wave * 16 + ln;
#pragma unroll
        for (int k = 0; k < 32; ++k) {
            const int K0 = 4 * k + 2 * g;
            v2f bb;
            bb.x = __expf(trans[(size_t)K0 * NN + col]);
            bb.y = __expf(trans[(size_t)(K0 + 1) * NN + col]);
            bfrag[k] = bb;
        }
    }

    for (int t = 1; t < TT; ++t) {
        // --- row max M_b = max_i alpha[b][i] ---
        {
            const int r = tid >> 4, s = tid & 15;
            const float* row = &alpha_s[r * STRIDE + s * 8];
            float mx = row[0];
#pragma unroll
            for (int q = 1; q < 8; ++q) mx = fmaxf(mx, row[q]);
            red_s[r * 16 + s] = mx;
        }
        __syncthreads();
        if (tid < 16) {
            float mx = red_s[tid * 16];
#pragma unroll
            for (int q = 1; q < 16; ++q) mx = fmaxf(mx, red_s[tid * 16 + q]);
            Mrow_s[tid] = mx;
        }
        __syncthreads();

        // --- expA = exp(alpha - M) ---
        for (int e = tid; e < 16 * NN; e += 256) {
            const int m = e >> 7, c = e & 127;
            expA_s[m * STRIDE + c] = __expf(alpha_s[m * STRIDE + c] - Mrow_s[m]);
        }
        __syncthreads();

        // --- GEMM: S = expA (16x128) x expT (128x16 tile), K=128 as 32 x (K=4) ---
        // A layout: lane ln holds row M=ln; VGPR v, half g -> K = 4k + v + 2g.
        v8f acc = {};
        const float* arow = &expA_s[ln * STRIDE + 2 * g];
#pragma unroll
        for (int k = 0; k < 32; ++k) {
            v2f a = *(const v2f*)(arow + 4 * k);
            acc = __builtin_amdgcn_wmma_f32_16x16x4_f32(
                false, a, false, bfrag[k], (short)0, acc, false, false);
        }

        // --- alpha' = valid ? log(S) + M + emit : alpha (own 16 cols only) ---
        const int col = wave * 16 + ln;
#pragma unroll
        for (int r = 0; r < 8; ++r) {
            const int m = r + 8 * g;                 // C/D layout: half g holds M = r + 8g
            const float oldv = alpha_s[m * STRIDE + col];
            const float emit = emissions[(size_t)(b0 + m) * TT * NN + (size_t)t * NN + col];
            const float nv   = __logf(acc[r]) + Mrow_s[m] + emit;
            alpha_s[m * STRIDE + col] = (t < ts_s[m]) ? nv : oldv;
        }
        __syncthreads();
    }

    // --- log_partitions = logsumexp_j(alpha + last) ---
    {
        const int r = tid >> 4, s = tid & 15;
        float mx = -INFINITY;
#pragma unroll
        for (int q = 0; q < 8; ++q) {
            const int c = s * 8 + q;
            mx = fmaxf(mx, alpha_s[r * STRIDE + c] + last[c]);
        }
        red_s[r * 16 + s] = mx;
        __syncthreads();
        if (tid < 16) {
            float m2 = red_s[tid * 16];
#pragma unroll
            for (int q = 1; q < 16; ++q) m2 = fmaxf(m2, red_s[tid * 16 + q]);
            Mrow_s[tid] = m2;
        }
        __syncthreads();
        float sm = 0.f;
#pragma unroll
        for (int q = 0; q < 8; ++q) {
            const int c = s * 8 + q;
            sm += __expf(alpha_s[r * STRIDE + c] + last[c] - Mrow_s[r]);
        }
        red_s[r * 16 + s] = sm;
        __syncthreads();
        if (tid < 16) {
            float ss = 0.f;
#pragma unroll
            for (int q = 0; q < 16; ++q) ss += red_s[tid * 16 + q];
            out[b0 + tid] = __logf(ss) + Mrow_s[tid];
        }
    }
}

// ---------------------------------------------------------------------------
// Gold-path score: one block per batch element.
// ---------------------------------------------------------------------------
__global__ __launch_bounds__(256) void crf_score_kernel(
    const float* __restrict__ emissions,
    const int*   __restrict__ token_sizes,
    const int*   __restrict__ targets,        // [64][1024]
    const float* __restrict__ trans,
    const float* __restrict__ head,
    const float* __restrict__ last,
    float* __restrict__ out)
{
    __shared__ float red[256];
    const int b = blockIdx.x, tid = threadIdx.x;
    const int ts = token_sizes[b];
    const int* tgt = &targets[(size_t)b * TT];

    float sum = 0.f;
    for (int t = tid; t < TT; t += 256) {
        if (t < ts) {
            const int ct = tgt[t];
            sum += emissions[(size_t)b * TT * NN + (size_t)t * NN + ct];
            if (t >= 1) sum += trans[(size_t)tgt[t - 1] * NN + ct];
        }
    }
    red[tid] = sum;
    __syncthreads();
    for (int off = 128; off > 0; off >>= 1) {
        if (tid < off) red[tid] += red[tid + off];
        __syncthreads();
    }
    if (tid == 0) {
        out[64 + b] = red[0] + head[tgt[0]] + last[tgt[ts - 1]];
    }
}

extern "C" void kernel_launch(void* const* d_in, const int* in_sizes, int n_in,
                              void* d_out, int out_size, void* d_ws, size_t ws_size,
                              hipStream_t stream) {
    const float* emissions = (const float*)d_in[0];
    const int*   tok_sz    = (const int*)  d_in[1];
    const int*   targets   = (const int*)  d_in[2];
    const float* trans     = (const float*)d_in[3];
    const float* head      = (const float*)d_in[4];
    const float* last      = (const float*)d_in[5];
    float* out = (float*)d_out;

    crf_scan_kernel<<<4, 256, 0, stream>>>(emissions, tok_sz, trans, head, last, out);
    crf_score_kernel<<<64, 256, 0, stream>>>(emissions, tok_sz, targets, trans, head, last, out);
}